// PerParticleDeterminantAntiequivarianceLayer_14851996910247
// MI455X (gfx1250) — compile-verified
//
#include <hip/hip_runtime.h>
#include <math.h>

// ---------------------------------------------------------------------------
// Per-particle determinant antiequivariance layer, fused single-pass kernel.
//
// MI455X reasoning: ~2.3 GFLOP vs ~262 MB HBM traffic -> pure bandwidth bound
// (~11us floor @ 23.3 TB/s). One wave32 per (b, spin) tile, f32 end-to-end.
//  - Two 16x512x16 GEMMs via V_WMMA_F32_16X16X4_F32 (C tile == one wave acc).
//  - Blocks are spin-uniform (8 waves, 8 consecutive b): the spin's 64KB of
//    W is staged into LDS ONCE per block by the Tensor Data Mover
//    (tensor_load_to_lds + s_wait_tensorcnt), so the WMMA B operand streams
//    from LDS instead of each wave pulling 64KB through the cache hierarchy
//    (cuts W read traffic from ~256MB to ~32MB of L2 reads).
//  - 16 determinants per wave via register-resident column-per-lane LU,
//    two matrices per pass (one per half-wave), shfl pivot broadcasts.
//  - Output = x * det with float4 streaming (x re-read hits L2).
// ---------------------------------------------------------------------------

typedef float        v2f __attribute__((ext_vector_type(2)));
typedef float        v8f __attribute__((ext_vector_type(8)));
typedef unsigned int v4u __attribute__((ext_vector_type(4)));
typedef int          v4i __attribute__((ext_vector_type(4)));
typedef int          v8i __attribute__((ext_vector_type(8)));

#define BB    2048
#define NELEC 32
#define DD    512
#define NION  8
#define NSPIN 2
#define NPART 16          // NELEC / NSPIN
#define WAVES_PER_BLOCK 8
#define W_ELEMS (2 * DD * NPART)   // 16384 f32 = 64KB per spin (W1 ++ W2)

__global__ __launch_bounds__(WAVES_PER_BLOCK * 32)
void ppdet_antiequi_kernel(const float* __restrict__ eq,     // (B, 32, 512)
                           const float* __restrict__ r_ei,   // (B, 32, 8, 3)
                           const float* __restrict__ W_lin,  // (2, 1024, 16)
                           const float* __restrict__ b_lin,  // (2, 16)
                           const float* __restrict__ A_env,  // (2, 3, 3, 8, 16)
                           const float* __restrict__ w_ion,  // (2, 8, 16)
                           float* __restrict__ out)          // (B, 2, 16, 512)
{
    __shared__ float sW[W_ELEMS];   // 64KB: this block's spin W1 (rows 0..511)
                                    //        then W2 (rows 512..1023), 16 cols

    const int lane = threadIdx.x & 31;
    const int wave = threadIdx.x >> 5;
    const int s    = blockIdx.x & 1;                         // spin-uniform block
    const int b    = (blockIdx.x >> 1) * WAVES_PER_BLOCK + wave;

    const int n = lane & 15;   // A-row / B-col / C-col index for this lane
    const int h = lane >> 4;   // half-wave select

    const float* wsrc = W_lin + (size_t)s * W_ELEMS;

    // ---------------- Phase 0: TDM-stage W[s] (64KB) into LDS --------------
#if __has_builtin(__builtin_amdgcn_tensor_load_to_lds)
    if (wave == 0) {
        const unsigned long long ga = (unsigned long long)(uintptr_t)wsrc;
        const unsigned int ldsA = (unsigned int)(uintptr_t)(&sW[0]); // LDS offset = low 32 bits of flat shared addr

        v4u g0;
        g0.x = 0x1u;                                   // count=1 valid user D#
        g0.y = ldsA;                                   // lds_addr (bytes)
        g0.z = (unsigned int)(ga & 0xffffffffu);       // global_addr[31:0]
        g0.w = (unsigned int)((ga >> 32) & 0x01ffffffu)// global_addr[56:32]
             | (2u << 30);                             // type=2 ("image")

        // 2D tensor, data_size=4B, one tile of 16384 contiguous elements.
        v8i g1;
        g1[0] = (int)(2u << 16);                       // wg_mask=0, data_size=2(4B)
        g1[1] = (int)((16384u & 0xffffu) << 16);       // tensor_dim0[15:0]
        g1[2] = (int)(1u << 16);                       // tensor_dim0[31:16]=0, tensor_dim1=1
        g1[3] = (int)(16384u << 16);                   // tile_dim0 = 16384
        g1[4] = 1;                                     // tile_dim1 = 1, tile_dim2 = 0
        g1[5] = 16384;                                 // tensor_dim0_stride[31:0]
        g1[6] = (int)((16384u & 0xffffu) << 16);       // stride[47:32]=0, dim1_stride[15:0]
        g1[7] = 0;                                     // dim1_stride[47:16]

        v4i g2 = {0, 0, 0, 0};                         // 2D: groups 2/3 unused
        v4i g3 = {0, 0, 0, 0};
#if defined(__clang_major__) && (__clang_major__ >= 23)
        v8i gz = {0, 0, 0, 0, 0, 0, 0, 0};
        __builtin_amdgcn_tensor_load_to_lds(g0, g1, g2, g3, gz, 0);
#else
        __builtin_amdgcn_tensor_load_to_lds(g0, g1, g2, g3, 0);
#endif
        __builtin_amdgcn_s_wait_tensorcnt(0);
    }
#else
    // Fallback: cooperative wide copy.
    for (int idx = threadIdx.x; idx < W_ELEMS / 4; idx += WAVES_PER_BLOCK * 32)
        ((float4*)sW)[idx] = ((const float4*)wsrc)[idx];
#endif
    __syncthreads();

    const float* xb = eq + ((size_t)b * NELEC + s * NPART) * DD;   // 16 x 512

    // ---------------- Phase 1: lin_p = x@W1, lin_j = x@W2 via f32 WMMA ------
    // A (16x4 f32) layout: lane 16h+m holds {A[m, k0+2h], A[m, k0+2h+1]}.
    // B (4x16 f32) layout: lane 16h+n holds {B[k0+2h, n], B[k0+2h+1, n]}.
    v8f c1 = {0.f, 0.f, 0.f, 0.f, 0.f, 0.f, 0.f, 0.f};  // lin_p
    v8f c2 = {0.f, 0.f, 0.f, 0.f, 0.f, 0.f, 0.f, 0.f};  // lin_j (pre-bias)

    #pragma unroll 4
    for (int k0 = 0; k0 < DD; k0 += 4) {
        const int ka = k0 + 2 * h;
        v2f a, b1v, b2v;
        const float* ap = xb + (size_t)n * DD + ka;   // row n of x, b64 load
        a.x   = ap[0];
        a.y   = ap[1];
        b1v.x = sW[ka * NPART + n];                   // W1 from LDS
        b1v.y = sW[(ka + 1) * NPART + n];
        b2v.x = sW[(ka + DD) * NPART + n];            // W2 from LDS
        b2v.y = sW[(ka + DD + 1) * NPART + n];
        c1 = __builtin_amdgcn_wmma_f32_16x16x4_f32(false, a, false, b1v,
                                                   (short)0, c1, false, false);
        c2 = __builtin_amdgcn_wmma_f32_16x16x4_f32(false, a, false, b2v,
                                                   (short)0, c2, false, false);
    }
    // C layout: c1[v] = lin_p[v + 8h, n];  c2[v] = lin_j[v + 8h, n].

    // ---------------- Phase 2: envelopes ------------------------------------
    // env[j, k] = sum_i w_ion[s,i,k] * exp(-||r[j,i,:] @ A_env[s,:,:,i,k]||)
    // This lane computes env rows j = v + 8h for its column k = n.
    const float  bias  = b_lin[s * NPART + n];
    const float* Abase = A_env + (size_t)s * (3 * 3 * NION * NPART);
    const float* wbase = w_ion + (size_t)s * (NION * NPART);
    const float* rbase = r_ei + ((size_t)b * NELEC + s * NPART) * NION * 3;

    float e_lo[8], u_lo[8];
    #pragma unroll
    for (int v = 0; v < 8; ++v) e_lo[v] = 0.f;

    #pragma unroll
    for (int i = 0; i < NION; ++i) {
        // 3x3 matrix for (i, k): independent of j, hoisted out of row loop.
        float A00 = Abase[((0 * 3 + 0) * NION + i) * NPART + n];
        float A01 = Abase[((0 * 3 + 1) * NION + i) * NPART + n];
        float A02 = Abase[((0 * 3 + 2) * NION + i) * NPART + n];
        float A10 = Abase[((1 * 3 + 0) * NION + i) * NPART + n];
        float A11 = Abase[((1 * 3 + 1) * NION + i) * NPART + n];
        float A12 = Abase[((1 * 3 + 2) * NION + i) * NPART + n];
        float A20 = Abase[((2 * 3 + 0) * NION + i) * NPART + n];
        float A21 = Abase[((2 * 3 + 1) * NION + i) * NPART + n];
        float A22 = Abase[((2 * 3 + 2) * NION + i) * NPART + n];
        float wi  = wbase[i * NPART + n];
        #pragma unroll
        for (int v = 0; v < 8; ++v) {
            const int j = v + 8 * h;
            const float* rp = rbase + ((size_t)j * NION + i) * 3;
            float r0 = rp[0], r1 = rp[1], r2 = rp[2];
            float a0 = fmaf(r0, A00, fmaf(r1, A10, r2 * A20));
            float a1 = fmaf(r0, A01, fmaf(r1, A11, r2 * A21));
            float a2 = fmaf(r0, A02, fmaf(r1, A12, r2 * A22));
            float nr = sqrtf(fmaf(a0, a0, fmaf(a1, a1, a2 * a2)));
            e_lo[v] = fmaf(wi, __expf(-nr), e_lo[v]);
        }
    }
    #pragma unroll
    for (int v = 0; v < 8; ++v)
        u_lo[v] = (c2[v] + bias) * e_lo[v];   // U[j,k] = lin_j[j,k]*env[j,k]

    // ---------------- Phase 3: gather full columns across half-waves --------
    // After this, lane 16h+k holds full column k (rows 0..15) of U and E.
    float uf[16], ef[16];
    #pragma unroll
    for (int v = 0; v < 8; ++v) {
        float uo = __shfl(u_lo[v], lane ^ 16);
        float eo = __shfl(e_lo[v], lane ^ 16);
        uf[v]     = h ? uo      : u_lo[v];
        uf[v + 8] = h ? u_lo[v] : uo;
        ef[v]     = h ? eo      : e_lo[v];
        ef[v + 8] = h ? e_lo[v] : eo;
    }

    // ---------------- Phase 4: 16 determinants, 2 per pass ------------------
    // Pass t: low half lanes -> particle t, high half -> particle t+8.
    // lin_p[p, k] for this lane's (p, k) is exactly the LOCAL c1[t].
    // Column-per-lane Gaussian elimination, pivot column broadcast by shfl.
    float dts[8];
    const int srcBase = lane & 16;
    #pragma unroll
    for (int t = 0; t < 8; ++t) {
        const float ap = c1[t];
        float mcol[16];
        #pragma unroll
        for (int r = 0; r < 16; ++r) mcol[r] = fmaf(ap, ef[r], uf[r]);

        float det = 1.f;
        #pragma unroll
        for (int sp = 0; sp < 16; ++sp) {
            float piv = __shfl(mcol[sp], sp + srcBase);
            det *= piv;
            float inv = 1.f / piv;
            #pragma unroll
            for (int r = 0; r < 16; ++r) {
                if (r > sp) {
                    float vr = __shfl(mcol[r], sp + srcBase);
                    mcol[r] = fmaf(-vr * inv, mcol[sp], mcol[r]);
                }
            }
        }
        dts[t] = det;   // particle (t + 8h), identical across a half-wave
    }

    // ---------------- Phase 5: out[p,:] = x[p,:] * det_p (streaming) --------
    float* outb = out + ((size_t)b * NELEC + s * NPART) * DD;
    #pragma unroll
    for (int p = 0; p < NPART; ++p) {
        const float det = __shfl(dts[p & 7], (p >> 3) << 4);
        const float4* xi = (const float4*)(xb + (size_t)p * DD);
        float4*       yo = (float4*)(outb + (size_t)p * DD);
        #pragma unroll
        for (int it = 0; it < (DD / 4) / 32; ++it) {
            const int idx = it * 32 + lane;
            float4 xv = xi[idx];
            xv.x *= det; xv.y *= det; xv.z *= det; xv.w *= det;
            yo[idx] = xv;
        }
    }
}

extern "C" void kernel_launch(void* const* d_in, const int* in_sizes, int n_in,
                              void* d_out, int out_size, void* d_ws, size_t ws_size,
                              hipStream_t stream) {
    (void)in_sizes; (void)n_in; (void)d_ws; (void)ws_size; (void)out_size;
    const float* eq    = (const float*)d_in[0];
    const float* r_ei  = (const float*)d_in[1];
    const float* W_lin = (const float*)d_in[2];
    const float* b_lin = (const float*)d_in[3];
    const float* A_env = (const float*)d_in[4];
    const float* w_ion = (const float*)d_in[5];
    float* out = (float*)d_out;

    const int tiles  = BB * NSPIN;                       // 4096 wave tiles
    const int blocks = tiles / WAVES_PER_BLOCK;          // 512 blocks of 256
    ppdet_antiequi_kernel<<<blocks, WAVES_PER_BLOCK * 32, 0, stream>>>(
        eq, r_ei, W_lin, b_lin, A_env, w_ion, out);
}